// SpatialTransformation_15814069584023
// MI455X (gfx1250) — compile-verified
//
#include <hip/hip_runtime.h>

// Trilinear 3D warp (grid sample with zero border pad=1), MI455X/gfx1250.
// Memory-bound gather: no WMMA applies (under 1 FLOP/byte). Optimized for the
// memory system instead: D-contiguous coalescing, NT hints on streamed data,
// global_prefetch_b8 to warm L2 at the identity-mapped neighborhood.

namespace {
constexpr int NB = 2, NC = 2, NH = 160, NW = 160, ND = 160;
constexpr long HWD = (long)NH * NW * ND;   // 4,096,000 elements per channel
constexpr int  WD  = NW * ND;              // 25,600 (fits 32-bit offsets)
}

__device__ __forceinline__ int clampi(int v, int lo, int hi) {
  return v < lo ? lo : (v > hi ? hi : v);
}

__global__ __launch_bounds__(ND) void warp3d_kernel(
    const float* __restrict__ img,   // [B,C,H,W,D]
    const float* __restrict__ dm,    // [B,3,H,W,D]
    float* __restrict__ out)         // [B,C,H,W,D]
{
  const int d = threadIdx.x;   // 160 lanes = 5 wave32s, stride-1 along D
  const int w = blockIdx.x;
  const int h = blockIdx.y;
  const int b = blockIdx.z;

  const int sp = h * WD + w * ND + d;               // spatial offset (32-bit)
  const float* dmb = dm + (long)b * 3 * HWD + sp;

  // Deformation field is streamed exactly once -> non-temporal (TH=NT),
  // keep L2 capacity for the reused image volume.
  const float mx = __builtin_nontemporal_load(dmb);
  const float my = __builtin_nontemporal_load(dmb + HWD);
  const float mz = __builtin_nontemporal_load(dmb + 2 * HWD);

  const float* imgb = img + (long)b * NC * HWD;

  // Displacements ~N(0,1): gathers cluster around the identity mapping.
  // Warm L2 ahead of the data-dependent loads (gfx1250 global_prefetch_b8).
  __builtin_prefetch(imgb + sp, 0, 1);
  __builtin_prefetch(imgb + HWD + sp, 0, 1);

  // Coordinates in the zero-padded volume (pad 1, then +1 as in reference).
  const float x = mx + (float)h + 1.0f;
  const float y = my + (float)w + 1.0f;
  const float z = mz + (float)d + 1.0f;

  const int fx = (int)floorf(x);
  const int fy = (int)floorf(y);
  const int fz = (int)floorf(z);

  // Reference clip order: x1 = clip(floor+1), x0 = clip(floor); weights use
  // the *clipped* x1.
  const int x1 = clampi(fx + 1, 0, NH + 1);
  const int y1 = clampi(fy + 1, 0, NW + 1);
  const int z1 = clampi(fz + 1, 0, ND + 1);
  const int x0 = clampi(fx, 0, NH + 1);
  const int y0 = clampi(fy, 0, NW + 1);
  const int z0 = clampi(fz, 0, ND + 1);

  const float dx = (float)x1 - x;
  const float dy = (float)y1 - y;
  const float dz = (float)z1 - z;
  const float ex = 1.0f - dx, ey = 1.0f - dy, ez = 1.0f - dz;

  // Padded index i maps to original i-1; padded border (i==0 or i==N+1) is 0.
  const bool vx0 = (x0 >= 1) & (x0 <= NH);
  const bool vx1 = (x1 >= 1) & (x1 <= NH);
  const bool vy0 = (y0 >= 1) & (y0 <= NW);
  const bool vy1 = (y1 >= 1) & (y1 <= NW);
  const bool vz0 = (z0 >= 1) & (z0 <= ND);
  const bool vz1 = (z1 >= 1) & (z1 <= ND);

  // Clamped in-bounds addresses (loads always legal; border handled by
  // zeroing the weight, which reproduces gather-of-zero exactly).
  const int ax0 = clampi(x0 - 1, 0, NH - 1);
  const int ax1 = clampi(x1 - 1, 0, NH - 1);
  const int ay0 = clampi(y0 - 1, 0, NW - 1);
  const int ay1 = clampi(y1 - 1, 0, NW - 1);
  const int az0 = clampi(z0 - 1, 0, ND - 1);
  const int az1 = clampi(z1 - 1, 0, ND - 1);

  const int r00 = ax0 * WD + ay0 * ND;
  const int r01 = ax0 * WD + ay1 * ND;
  const int r10 = ax1 * WD + ay0 * ND;
  const int r11 = ax1 * WD + ay1 * ND;

  const int o000 = r00 + az0, o001 = r00 + az1;
  const int o010 = r01 + az0, o011 = r01 + az1;
  const int o100 = r10 + az0, o101 = r10 + az1;
  const int o110 = r11 + az0, o111 = r11 + az1;

  const float w000 = (vx0 & vy0 & vz0) ? dx * dy * dz : 0.0f;
  const float w001 = (vx0 & vy0 & vz1) ? dx * dy * ez : 0.0f;
  const float w010 = (vx0 & vy1 & vz0) ? dx * ey * dz : 0.0f;
  const float w011 = (vx0 & vy1 & vz1) ? dx * ey * ez : 0.0f;
  const float w100 = (vx1 & vy0 & vz0) ? ex * dy * dz : 0.0f;
  const float w101 = (vx1 & vy0 & vz1) ? ex * dy * ez : 0.0f;
  const float w110 = (vx1 & vy1 & vz0) ? ex * ey * dz : 0.0f;
  const float w111 = (vx1 & vy1 & vz1) ? ex * ey * ez : 0.0f;

  float* outb = out + (long)b * NC * HWD + sp;
#pragma unroll
  for (int c = 0; c < NC; ++c) {
    const float* p = imgb + (long)c * HWD;
    float acc = w000 * p[o000];          // same left-to-right order as ref
    acc += w001 * p[o001];
    acc += w010 * p[o010];
    acc += w011 * p[o011];
    acc += w100 * p[o100];
    acc += w101 * p[o101];
    acc += w110 * p[o110];
    acc += w111 * p[o111];
    // Output is write-once, never re-read -> non-temporal store.
    __builtin_nontemporal_store(acc, outb + (long)c * HWD);
  }
}

extern "C" void kernel_launch(void* const* d_in, const int* in_sizes, int n_in,
                              void* d_out, int out_size, void* d_ws, size_t ws_size,
                              hipStream_t stream) {
  (void)in_sizes; (void)n_in; (void)out_size; (void)d_ws; (void)ws_size;
  const float* img = (const float*)d_in[0];   // moving_image [2,2,160,160,160]
  const float* dm  = (const float*)d_in[1];   // deformation  [2,3,160,160,160]
  float* out = (float*)d_out;                 // [2,2,160,160,160]

  dim3 grid(NW, NH, NB);   // one block per (b,h,w)
  dim3 block(ND);          // d along threadIdx.x -> fully coalesced
  warp3d_kernel<<<grid, block, 0, stream>>>(img, dm, out);
}